// GNN_AA_1846835937281
// MI455X (gfx1250) — compile-verified
//
#include <hip/hip_runtime.h>
#include <math.h>

// ---------------------------------------------------------------------------
// MI455X / gfx1250, wave32. GNN edge aggregation + two masked cosine-attention
// blocks. Attention GEMMs: V_WMMA_F32_16X16X4_F32 (exact f32; K=64 -> 16
// steps). Round 3: round 2's full operand preload spilled to scratch (spill
// reloads serialized every WMMA). This version caps register pressure:
// A tile resident (32 VGPRs), B frags staged 8-at-a-time, second-GEMM B frags
// 4-at-a-time, dual sim accumulation chains, adj preload + global_prefetch.
// ---------------------------------------------------------------------------

typedef __attribute__((ext_vector_type(2))) float v2f;
typedef __attribute__((ext_vector_type(8))) float v8f;

#define DIMN   64
#define NNODES 100000
#define NUSERS 4096
#define NITEMS 8192
#define NEGC   (-9.0e15f)
#define WPB    4            // waves per block in attention kernel
#define RECF   1056         // partial record: 16 m + 16 s + 16*64 acc

static __device__ __forceinline__ float waveReduceSum(float v) {
#pragma unroll
  for (int m = 16; m >= 1; m >>= 1) v += __shfl_xor(v, m, 32);
  return v;
}
// reductions within a 16-lane half (xor masks 1..8 never cross the half)
static __device__ __forceinline__ float halfReduceMax(float v) {
#pragma unroll
  for (int m = 8; m >= 1; m >>= 1) v = fmaxf(v, __shfl_xor(v, m, 32));
  return v;
}
static __device__ __forceinline__ float halfReduceSum(float v) {
#pragma unroll
  for (int m = 8; m >= 1; m >>= 1) v += __shfl_xor(v, m, 32);
  return v;
}

// ---------------------------------------------------------------------------
// Row L2-normalize (F.normalize p=2 dim=1 eps=1e-12). One wave per 64-f row.
// ---------------------------------------------------------------------------
__global__ void rownorm_kernel(const float* __restrict__ in,
                               float* __restrict__ out, int rows) {
  const int lane = threadIdx.x & 31;
  const int wid  = (blockIdx.x * blockDim.x + threadIdx.x) >> 5;
  const int nw   = (gridDim.x * blockDim.x) >> 5;
  for (int r = wid; r < rows; r += nw) {
    v2f v = *(const v2f*)(in + (size_t)r * DIMN + 2 * lane);
    float ss = waveReduceSum(v[0] * v[0] + v[1] * v[1]);
    float scale = 1.0f / fmaxf(sqrtf(ss), 1e-12f);
    v2f o; o[0] = v[0] * scale; o[1] = v[1] * scale;
    *(v2f*)(out + (size_t)r * DIMN + 2 * lane) = o;
  }
}

// ---------------------------------------------------------------------------
// inner_GNN: one wave per edge (grid-stride). 2 floats/lane of the 64-dim row.
// gate MLP: lane j handles hidden units j and j+32; two wave reductions.
// segment_sum via global_atomic_add_f32 into pre-zeroed aggr.
// ---------------------------------------------------------------------------
__global__ void gnn_edge_kernel(const float* __restrict__ x,
                                const int* __restrict__ ei,
                                const float* __restrict__ w1,
                                const float* __restrict__ b1,
                                const float* __restrict__ w2,
                                const float* __restrict__ b2,
                                float* __restrict__ aggr, int E) {
  const int lane = threadIdx.x & 31;
  const int wid  = (blockIdx.x * blockDim.x + threadIdx.x) >> 5;
  const int nw   = (gridDim.x * blockDim.x) >> 5;
  const float w1a = w1[lane], w1b = w1[lane + 32];
  const float b1a = b1[lane], b1b = b1[lane + 32];
  const float w2a = w2[lane], w2b = w2[lane + 32];
  const float bb2 = b2[0];
  for (int e = wid; e < E; e += nw) {
    const int s = ei[e];          // edge_index[0] = source (x_j)
    const int d = ei[E + e];      // edge_index[1] = target (aggregation idx)
    v2f xs = *(const v2f*)(x + (size_t)s * DIMN + 2 * lane);
    v2f xd = *(const v2f*)(x + (size_t)d * DIMN + 2 * lane);
    float p0 = xs[0] * xd[0], p1 = xs[1] * xd[1];
    float g = waveReduceSum(p0 + p1) * (1.0f / 64.0f);    // mean pool
    float h0 = fmaxf(g * w1a + b1a, 0.0f);                // relu(lin1)
    float h1 = fmaxf(g * w1b + b1b, 0.0f);
    float z = waveReduceSum(h0 * w2a + h1 * w2b) + bb2;   // lin2
    float c = 1.0f + 1.0f / (1.0f + __expf(-z));          // 1 + sigmoid
    float* dst = aggr + (size_t)d * DIMN + 2 * lane;
    atomicAdd(dst, c * p0);
    atomicAdd(dst + 1, c * p1);
  }
}

// ---------------------------------------------------------------------------
// Fused masked-softmax attention, flash style, WMMA f32 16x16x4.
// One wave owns 16 rows x a column segment; partial (m,s,acc) -> workspace.
// Layouts (CDNA5 ISA):
//   A 16x4 f32: lane = M%16, VGPR {0,1} = K {0,1} (lanes<16) / K {2,3} (>=16)
//   B 4x16 f32: lane = N,    VGPR {0,1} = K {0,1} / {2,3} by lane half
//   C/D 16x16:  VGPR r = row r (lanes 0-15) / row r+8 (lanes 16-31), lane = N
// ---------------------------------------------------------------------------
__global__ void __launch_bounds__(WPB * 32)
att_partial_kernel(const float* __restrict__ normA,  // R x 64
                   const float* __restrict__ normB,  // C x 64
                   const float* __restrict__ embB,   // C x 64 (raw)
                   const float* __restrict__ adj,    // R x C
                   float* __restrict__ partial,
                   int R, int C, int split) {
  __shared__ float plds[WPB][16 * 17];
  const int lane  = threadIdx.x & 31;
  const int wsl   = threadIdx.x >> 5;
  const int gw    = blockIdx.x * WPB + wsl;
  const int tiles = R >> 4;
  if (gw >= tiles * split) return;
  const int tile = gw / split;
  const int seg  = gw - tile * split;
  const int segC = C / split;
  const int c0 = seg * segC, c1 = c0 + segC;
  const int hi = lane >> 4;        // lane half
  const int cl = lane & 15;        // lane within half
  const int r0 = tile << 4;
  float* P = &plds[wsl][0];

  // Keep whole 16x64 A tile in registers: 16 K-chunks of 16x4 (32 VGPRs).
  v2f afrag[16];
#pragma unroll
  for (int kc = 0; kc < 16; ++kc)
    afrag[kc] = *(const v2f*)(normA + (size_t)(r0 + cl) * DIMN + 4 * kc + 2 * hi);

  v8f acc0 = {}, acc1 = {}, acc2 = {}, acc3 = {};
  float m[8], s[8];
#pragma unroll
  for (int r = 0; r < 8; ++r) { m[r] = -INFINITY; s[r] = 0.0f; }

  // Pointer-stepped bases (in-loop offsets become load immediates).
  const float* bPtr = normB + (size_t)(c0 + cl) * DIMN + 2 * hi;      // += 16*64
  const float* ePtr = embB + (size_t)c0 * DIMN + 2 * hi * DIMN + cl;  // += 16*64
  const float* adjP = adj + (size_t)(r0 + 8 * hi) * C + c0 + cl;      // += 16
  int adjOff[8];
#pragma unroll
  for (int r = 0; r < 8; ++r) adjOff[r] = r * C;

  for (int col = c0; col < c1; col += 16) {
    // adj tile values (C/D layout) — only HBM-resident stream; issue first.
    float av[8];
#pragma unroll
    for (int r = 0; r < 8; ++r) av[r] = adjP[adjOff[r]];
    // prefetch adj 4 chunks (256B) ahead to keep the HBM stream deep
    if (col + 64 < c1) {
#pragma unroll
      for (int r = 0; r < 8; ++r)
        __builtin_prefetch(adjP + adjOff[r] + 64, 0, 3);
    }

    // ---- sim tile: S = A(16x64) @ normB_chunk^T(64x16) ----
    // B frags staged 8 at a time (16 VGPRs): batch of loads -> one wait,
    // two independent accumulation chains for WMMA ILP.
    v8f sv0 = {}, sv1 = {};
    {
      v2f bfr[8];
#pragma unroll
      for (int kc = 0; kc < 8; ++kc)
        bfr[kc] = *(const v2f*)(bPtr + 4 * kc);
#pragma unroll
      for (int kc = 0; kc < 4; ++kc) {
        sv0 = __builtin_amdgcn_wmma_f32_16x16x4_f32(false, afrag[2 * kc], false,
                                                    bfr[2 * kc], (short)0, sv0,
                                                    false, false);
        sv1 = __builtin_amdgcn_wmma_f32_16x16x4_f32(false, afrag[2 * kc + 1], false,
                                                    bfr[2 * kc + 1], (short)0, sv1,
                                                    false, false);
      }
#pragma unroll
      for (int kc = 0; kc < 8; ++kc)
        bfr[kc] = *(const v2f*)(bPtr + 32 + 4 * kc);
#pragma unroll
      for (int kc = 0; kc < 4; ++kc) {
        sv0 = __builtin_amdgcn_wmma_f32_16x16x4_f32(false, afrag[8 + 2 * kc], false,
                                                    bfr[2 * kc], (short)0, sv0,
                                                    false, false);
        sv1 = __builtin_amdgcn_wmma_f32_16x16x4_f32(false, afrag[9 + 2 * kc], false,
                                                    bfr[2 * kc + 1], (short)0, sv1,
                                                    false, false);
      }
    }

    // ---- adj mask + online softmax (per-row stats in each 16-lane half) ----
#pragma unroll
    for (int r = 0; r < 8; ++r) {
      float v = (sv0[r] + sv1[r]) * av[r];
      v = (v != 0.0f) ? v : NEGC;                  // jnp.where(sim!=0, sim, NEG)
      float cmax = halfReduceMax(v);
      float nm = fmaxf(m[r], cmax);
      float scale = __expf(m[r] - nm);             // first iter: exp(-inf)=0
      float p = __expf(v - nm);
      float rs = halfReduceSum(p);
      s[r] = s[r] * scale + rs;
      m[r] = nm;
      acc0[r] *= scale; acc1[r] *= scale; acc2[r] *= scale; acc3[r] *= scale;
      P[(r + 8 * hi) * 17 + cl] = p;               // C/D layout -> row-major LDS
    }

    // ---- acc(16x64) += P(16x16) @ embB_chunk(16x64): 4 N-tiles x 4 K-steps ---
    // Same-wave LDS ops are in-order; no barrier needed. B frags staged 4 per
    // K-step (8 VGPRs) so all 4 loads issue as one batch before their WMMAs.
#pragma unroll
    for (int kc = 0; kc < 4; ++kc) {
      v2f eb[4];
#pragma unroll
      for (int t = 0; t < 4; ++t) {
        v2f b;
        b[0] = ePtr[(4 * kc) * DIMN + 16 * t];
        b[1] = ePtr[(4 * kc + 1) * DIMN + 16 * t];
        eb[t] = b;
      }
      v2f pf;
      pf[0] = P[cl * 17 + 4 * kc + 2 * hi];
      pf[1] = P[cl * 17 + 4 * kc + 2 * hi + 1];
      acc0 = __builtin_amdgcn_wmma_f32_16x16x4_f32(false, pf, false, eb[0], (short)0, acc0, false, false);
      acc1 = __builtin_amdgcn_wmma_f32_16x16x4_f32(false, pf, false, eb[1], (short)0, acc1, false, false);
      acc2 = __builtin_amdgcn_wmma_f32_16x16x4_f32(false, pf, false, eb[2], (short)0, acc2, false, false);
      acc3 = __builtin_amdgcn_wmma_f32_16x16x4_f32(false, pf, false, eb[3], (short)0, acc3, false, false);
    }

    bPtr += 16 * DIMN;
    ePtr += 16 * DIMN;
    adjP += 16;
  }

  // ---- write partial record [16 m | 16 s | 16x64 acc] ----
  float* rec = partial + (size_t)gw * RECF;
  if (cl == 0) {
#pragma unroll
    for (int r = 0; r < 8; ++r) {
      rec[r + 8 * hi] = m[r];
      rec[16 + r + 8 * hi] = s[r];
    }
  }
#pragma unroll
  for (int r = 0; r < 8; ++r) {
    const int row = r + 8 * hi;
    rec[32 + row * 64 + cl]      = acc0[r];
    rec[32 + row * 64 + 16 + cl] = acc1[r];
    rec[32 + row * 64 + 32 + cl] = acc2[r];
    rec[32 + row * 64 + 48 + cl] = acc3[r];
  }
}

// ---------------------------------------------------------------------------
// Merge column-split softmax partials and normalize. One block per 16-row tile.
// ---------------------------------------------------------------------------
__global__ void att_combine_kernel(const float* __restrict__ partial,
                                   float* __restrict__ out, int split) {
  __shared__ float wgt[8][16];
  __shared__ float denom[16];
  const int tile = blockIdx.x;
  const int tid = threadIdx.x;
  if (tid < 16) {
    float gm = -INFINITY;
    for (int sg = 0; sg < split; ++sg)
      gm = fmaxf(gm, partial[(size_t)(tile * split + sg) * RECF + tid]);
    float den = 0.0f;
    for (int sg = 0; sg < split; ++sg) {
      const float* rec = partial + (size_t)(tile * split + sg) * RECF;
      float w = __expf(rec[tid] - gm);
      wgt[sg][tid] = w;
      den += w * rec[16 + tid];
    }
    denom[tid] = den;   // always > 0 (fully-masked rows sum exp(0)=1 per col)
  }
  __syncthreads();
  for (int idx = tid; idx < 1024; idx += blockDim.x) {
    const int r = idx >> 6;
    float a = 0.0f;
    for (int sg = 0; sg < split; ++sg)
      a += wgt[sg][r] * partial[(size_t)(tile * split + sg) * RECF + 32 + idx];
    out[(size_t)tile * 1024 + idx] = a / denom[r];
  }
}

// ---------------------------------------------------------------------------
extern "C" void kernel_launch(void* const* d_in, const int* in_sizes, int n_in,
                              void* d_out, int out_size, void* d_ws, size_t ws_size,
                              hipStream_t stream) {
  const float* x    = (const float*)d_in[0];
  const float* w1   = (const float*)d_in[1];
  const float* b1   = (const float*)d_in[2];
  const float* w2   = (const float*)d_in[3];
  const float* b2   = (const float*)d_in[4];
  const float* ue   = (const float*)d_in[5];
  const float* ie   = (const float*)d_in[6];
  const float* uadj = (const float*)d_in[7];
  const float* iadj = (const float*)d_in[8];
  const int*   ei   = (const int*)d_in[9];
  const int E = in_sizes[9] / 2;

  float* aggr = (float*)d_out;
  float* outU = aggr + (size_t)NNODES * DIMN;
  float* outI = outU + (size_t)NUSERS * DIMN;

  const int tilesU = NUSERS / 16, tilesI = NITEMS / 16;
  const size_t normFloats = (size_t)(NUSERS + NITEMS) * DIMN;

  // Largest column-split whose partial records fit in workspace (deterministic
  // given ws_size). split=8 -> ~29MB, 4 -> ~16MB, 1 -> ~6.4MB.
  int split = 1;
  const int cand[4] = {8, 4, 2, 1};
  for (int i = 0; i < 4; ++i) {
    size_t need = (normFloats + (size_t)(tilesU + tilesI) * cand[i] * RECF) * sizeof(float);
    if (need <= ws_size) { split = cand[i]; break; }
  }

  float* normU = (float*)d_ws;
  float* normI = normU + (size_t)NUSERS * DIMN;
  float* partU = normI + (size_t)NITEMS * DIMN;
  float* partI = partU + (size_t)tilesU * split * RECF;

  // segment_sum target must start at zero every call
  hipMemsetAsync(aggr, 0, (size_t)NNODES * DIMN * sizeof(float), stream);

  rownorm_kernel<<<(NUSERS + 3) / 4, 128, 0, stream>>>(ue, normU, NUSERS);
  rownorm_kernel<<<(NITEMS + 3) / 4, 128, 0, stream>>>(ie, normI, NITEMS);

  gnn_edge_kernel<<<2048, 256, 0, stream>>>(x, ei, w1, b1, w2, b2, aggr, E);

  {
    const int waves = tilesU * split;
    att_partial_kernel<<<(waves + WPB - 1) / WPB, 32 * WPB, 0, stream>>>(
        normU, normI, ie, uadj, partU, NUSERS, NITEMS, split);
  }
  {
    const int waves = tilesI * split;
    att_partial_kernel<<<(waves + WPB - 1) / WPB, 32 * WPB, 0, stream>>>(
        normI, normU, ue, iadj, partI, NITEMS, NUSERS, split);
  }
  att_combine_kernel<<<tilesU, 128, 0, stream>>>(partU, outU, split);
  att_combine_kernel<<<tilesI, 128, 0, stream>>>(partI, outI, split);
}